// ece_function_69630009803210
// MI455X (gfx1250) — compile-verified
//
#include <hip/hip_runtime.h>

// ECE binning: conf = max_c probs, pred = argmax_c probs, acc = (pred==label)
// bin = clip(ceil(conf*15)-1, 0, 14); conf<=0 -> overflow bucket 15 (dropped)
// outputs: count[15] ++ conf_sum[15] ++ acc_sum[15]  (45 floats)

#define NBINS    15
#define NSLOT    16          // 15 real bins + overflow bucket
#define SLOTS    48          // 3 stats * 16 slots
#define TSTRIDE  49          // padded per-thread LDS stride (odd -> bank-conflict-free)
#define THREADS  256
#define MAXBLOCKS 1024

__device__ __forceinline__ void accum_pixel(float* __restrict__ h,
                                            float conf, int pred, int lab) {
    // matches: idx = clip(ceil(conf*15)-1, 0, 14); where(conf>0, idx, 15)
    int b = (int)ceilf(conf * 15.0f) - 1;
    b = b < 0 ? 0 : (b > (NBINS - 1) ? (NBINS - 1) : b);
    b = (conf > 0.0f) ? b : NBINS;          // overflow bucket
    h[b]             += 1.0f;               // count
    h[NSLOT + b]     += conf;               // conf_sum
    h[2 * NSLOT + b] += (pred == lab) ? 1.0f : 0.0f;  // acc_sum
}

template <int C_STATIC>
__global__ __launch_bounds__(THREADS)
void ece_partials_kernel(const float* __restrict__ probs,
                         const int*   __restrict__ labels,
                         float*       __restrict__ partials,
                         int C_rt, int P) {
    const int C = (C_STATIC > 0) ? C_STATIC : C_rt;

    __shared__ float hist[THREADS * TSTRIDE];
    const int tid  = threadIdx.x;
    const int base = tid * TSTRIDE;
    float* __restrict__ h = &hist[base];
    #pragma unroll
    for (int i = 0; i < SLOTS; ++i) h[i] = 0.0f;
    // (no sync needed: regions are thread-private until the reduction)

    const int     P4      = P >> 2;                    // float4 groups
    const size_t  plane4  = (size_t)P4;                // class-plane stride in groups
    const float4* pr4     = reinterpret_cast<const float4*>(probs);
    const int4*   lb4     = reinterpret_cast<const int4*>(labels);
    const int     gstride = gridDim.x * blockDim.x;

    for (int g = blockIdx.x * blockDim.x + tid; g < P4; g += gstride) {
        if (g + gstride < P4) {
            // gfx1250 global_prefetch_b8 for the next grid-stride line
            __builtin_prefetch(&pr4[g + gstride], 0, 0);
        }
        float4 best = pr4[g];                          // class 0
        int bx = 0, by = 0, bz = 0, bw = 0;
        #pragma unroll
        for (int c = 1; c < C; ++c) {
            float4 v = pr4[(size_t)c * plane4 + g];    // global_load_b128 stream
            if (v.x > best.x) { best.x = v.x; bx = c; } // strict > keeps first max
            if (v.y > best.y) { best.y = v.y; by = c; }
            if (v.z > best.z) { best.z = v.z; bz = c; }
            if (v.w > best.w) { best.w = v.w; bw = c; }
        }
        int4 lab = lb4[g];
        accum_pixel(h, best.x, bx, lab.x);
        accum_pixel(h, best.y, by, lab.y);
        accum_pixel(h, best.z, bz, lab.z);
        accum_pixel(h, best.w, bw, lab.w);
    }

    // scalar tail if P not divisible by 4 (distributed over first threads of block 0)
    const int rem = P & 3;
    if (rem && blockIdx.x == 0 && tid < rem) {
        const int p = (P & ~3) + tid;
        float best = probs[p];
        int   bc   = 0;
        for (int c = 1; c < C; ++c) {
            float v = probs[(size_t)c * (size_t)P + p];
            if (v > best) { best = v; bc = c; }
        }
        accum_pixel(h, best, bc, labels[p]);
    }

    __syncthreads();

    // Deterministic block reduction: thread j (<48) sums slot j across all
    // 256 thread-private regions in fixed order. No atomics.
    if (tid < SLOTS) {
        float s = 0.0f;
        for (int t = 0; t < THREADS; ++t) s += hist[t * TSTRIDE + tid];
        partials[(size_t)blockIdx.x * SLOTS + tid] = s;
    }
}

__global__ __launch_bounds__(64)
void ece_finalize_kernel(const float* __restrict__ partials,
                         float* __restrict__ out, int nblocks) {
    const int tid = threadIdx.x;
    if (tid < 3 * NBINS) {                 // 45 outputs
        const int group = tid / NBINS;     // 0=count, 1=conf_sum, 2=acc_sum
        const int bin   = tid - group * NBINS;
        const int slot  = group * NSLOT + bin;   // skip overflow bucket
        float s = 0.0f;
        for (int b = 0; b < nblocks; ++b) s += partials[(size_t)b * SLOTS + slot];
        out[tid] = s;                      // fixed-order sum -> bit-deterministic
    }
}

extern "C" void kernel_launch(void* const* d_in, const int* in_sizes, int n_in,
                              void* d_out, int out_size, void* d_ws, size_t ws_size,
                              hipStream_t stream) {
    const float* probs  = (const float*)d_in[0];
    const int*   labels = (const int*)d_in[1];
    float*       out    = (float*)d_out;
    float*       parts  = (float*)d_ws;

    const int P = in_sizes[1];            // pixels (1*1*1024*2048)
    const int C = in_sizes[0] / P;        // classes (19)

    int nblocks = MAXBLOCKS;
    const size_t per_block = (size_t)SLOTS * sizeof(float);
    if ((size_t)nblocks * per_block > ws_size) {
        nblocks = (int)(ws_size / per_block);
        if (nblocks < 1) nblocks = 1;
    }

    if (C == 19) {
        ece_partials_kernel<19><<<nblocks, THREADS, 0, stream>>>(probs, labels, parts, C, P);
    } else {
        ece_partials_kernel<0><<<nblocks, THREADS, 0, stream>>>(probs, labels, parts, C, P);
    }
    ece_finalize_kernel<<<1, 64, 0, stream>>>(parts, out, nblocks);
}